// PerDateSpearmanLoss_5832565588517
// MI455X (gfx1250) — compile-verified
//
#include <hip/hip_runtime.h>
#include <hip/hip_bf16.h>
#include <stdint.h>

// ---------------------------------------------------------------------------
// Per-date Spearman-style loss: segmented 6-moment reduction + finalize.
// N = 16.7M elements, D <= 1024 dates. Memory-bound: 192MB stream ~8.2us at
// 23.3 TB/s. Per-block LDS accumulation table with wave32 ds_add_f32 atomics
// (records padded to 7 floats: odd dword stride => full 64-bank spread, vs
// 6-float records whose addresses 6d+k only reach even banks), drained via
// gfx1250 async LDS->global stores, then deterministic column reduction and
// a tiny finalize kernel. No global float atomics => bitwise deterministic.
// ---------------------------------------------------------------------------

#define DMAX    1024
#define REC     7                       // 6 stats + 1 pad (odd LDS stride)
#define TABLE   (DMAX * REC)            // 7168 floats = 28672 bytes = 7*4096
#define TPB     256

typedef float vf4 __attribute__((ext_vector_type(4)));
typedef int   vi4 __attribute__((ext_vector_type(4)));

__device__ __forceinline__ void accum6(float* __restrict__ acc, int d,
                                       float p, float t) {
    float* r = acc + d * REC;           // +k folds into ds offset field
    atomicAdd(r + 0, 1.0f);             // n
    atomicAdd(r + 1, p);                // sum_p
    atomicAdd(r + 2, t);                // sum_t
    atomicAdd(r + 3, p * p);            // sum_pp
    atomicAdd(r + 4, t * t);            // sum_tt
    atomicAdd(r + 5, p * t);            // sum_pt
}

__global__ __launch_bounds__(TPB)
void spearman_partial(const float* __restrict__ pred,
                      const float* __restrict__ targ,
                      const int*   __restrict__ dates,
                      float*       __restrict__ partials,
                      int n)
{
    __shared__ float acc[TABLE];
    const int tid = threadIdx.x;
    for (int i = tid; i < TABLE; i += TPB) acc[i] = 0.0f;
    __syncthreads();

    const int n4 = n >> 2;
    const vf4* __restrict__ p4 = (const vf4*)pred;
    const vf4* __restrict__ t4 = (const vf4*)targ;
    const vi4* __restrict__ d4 = (const vi4*)dates;
    const int stride = (int)gridDim.x * TPB;

    for (int i = (int)blockIdx.x * TPB + tid; i < n4; i += stride) {
        vf4 p = __builtin_nontemporal_load(&p4[i]);   // global_load_b128 nt
        vf4 t = __builtin_nontemporal_load(&t4[i]);
        vi4 d = __builtin_nontemporal_load(&d4[i]);
        accum6(acc, d.x, p.x, t.x);
        accum6(acc, d.y, p.y, t.y);
        accum6(acc, d.z, p.z, t.z);
        accum6(acc, d.w, p.w, t.w);
    }

    // Tail (n % 4), handled once by block 0.
    const int done = n4 << 2;
    const int rem  = n - done;
    if (blockIdx.x == 0 && tid < rem) {
        int i = done + tid;
        accum6(acc, dates[i], pred[i], targ[i]);
    }
    __syncthreads();

    // Drain the 28KB LDS table straight to global memory via the CDNA5
    // async LDS->global path (no VGPR bounce). 256 lanes x 16B = 4KB per
    // instruction -> 7 instructions cover the table.
    float*   row      = partials + (size_t)blockIdx.x * TABLE;
    uint32_t lds_base = (uint32_t)(uintptr_t)(&acc[0]);
#pragma unroll
    for (int it = 0; it < (TABLE * 4) / (TPB * 16); ++it) {   // 7 iterations
        uint32_t off = (uint32_t)(tid * 16 + it * (TPB * 16));
        uint64_t g   = (uint64_t)(uintptr_t)row + off;
        uint32_t l   = lds_base + off;
        asm volatile("global_store_async_from_lds_b128 %0, %1, off"
                     :: "v"(g), "v"(l) : "memory");
    }
    asm volatile("s_wait_asynccnt 0x0" ::: "memory");
}

// Deterministic reduction of B partial rows -> one 7168-float table.
__global__ __launch_bounds__(TPB)
void spearman_colreduce(const float* __restrict__ partials,
                        float*       __restrict__ reduced,
                        int B)
{
    const int idx = blockIdx.x * TPB + threadIdx.x;
    if (idx >= TABLE) return;
    float s = 0.0f;
    int b = 0;
    for (; b + 4 <= B; b += 4) {
        s += partials[(size_t)(b + 0) * TABLE + idx];
        s += partials[(size_t)(b + 1) * TABLE + idx];
        s += partials[(size_t)(b + 2) * TABLE + idx];
        s += partials[(size_t)(b + 3) * TABLE + idx];
    }
    for (; b < B; ++b) s += partials[(size_t)b * TABLE + idx];
    reduced[idx] = s;
}

// Per-date correlation + averaging. Dates beyond the real num_dates have
// n == 0 -> excluded by the n >= 2 validity test, matching the reference.
__global__ __launch_bounds__(TPB)
void spearman_finalize(const float* __restrict__ red,
                       float*       __restrict__ out)
{
    __shared__ float s_c[TPB];
    __shared__ float s_v[TPB];
    float csum = 0.0f, vcnt = 0.0f;
    for (int d = threadIdx.x; d < DMAX; d += TPB) {
        const float* r = &red[d * REC];
        float nn = r[0];
        if (nn >= 2.0f) {
            float sp = r[1], st = r[2], spp = r[3], stt = r[4], spt = r[5];
            float inv = 1.0f / nn;
            float ssp = fmaxf(spp - sp * sp * inv, 0.0f);
            float sst = fmaxf(stt - st * st * inv, 0.0f);
            float cpt = spt - sp * st * inv;
            float dof = fmaxf(nn - 1.0f, 1.0f);
            float stdp = sqrtf(ssp / dof) + 1e-8f;
            float stdt = sqrtf(sst / dof) + 1e-8f;
            float c = (cpt * inv) / (stdp * stdt);
            c = fminf(1.0f, fmaxf(-1.0f, c));
            csum += c;
            vcnt += 1.0f;
        }
    }
    s_c[threadIdx.x] = csum;
    s_v[threadIdx.x] = vcnt;
    __syncthreads();
    for (int off = TPB / 2; off > 0; off >>= 1) {
        if (threadIdx.x < off) {
            s_c[threadIdx.x] += s_c[threadIdx.x + off];
            s_v[threadIdx.x] += s_v[threadIdx.x + off];
        }
        __syncthreads();
    }
    if (threadIdx.x == 0) out[0] = 1.0f - s_c[0] / fmaxf(s_v[0], 1.0f);
}

extern "C" void kernel_launch(void* const* d_in, const int* in_sizes, int n_in,
                              void* d_out, int out_size, void* d_ws, size_t ws_size,
                              hipStream_t stream)
{
    const float* pred  = (const float*)d_in[0];
    const float* targ  = (const float*)d_in[1];
    const int*   dates = (const int*)d_in[2];
    const int    n     = in_sizes[0];
    float* out = (float*)d_out;

    // ws layout: [0, 28672) reduced table; [28672, ...) B partial rows.
    const size_t row_bytes = (size_t)TABLE * sizeof(float);
    float* reduced  = (float*)d_ws;
    float* partials = (float*)((char*)d_ws + row_bytes);

    size_t avail = (ws_size > row_bytes) ? (ws_size - row_bytes) : 0;
    int B = (int)(avail / row_bytes);
    if (B > 768) B = 768;
    if (B < 1)   B = 1;

    spearman_partial<<<B, TPB, 0, stream>>>(pred, targ, dates, partials, n);
    spearman_colreduce<<<(TABLE + TPB - 1) / TPB, TPB, 0, stream>>>(partials, reduced, B);
    spearman_finalize<<<1, TPB, 0, stream>>>(reduced, out);
}